// GraphAtrousConv_26414048870630
// MI455X (gfx1250) — compile-verified
//
#include <hip/hip_runtime.h>
#include <hip/hip_bf16.h>

#define Bn 16384
#define Vn 25
#define Cn 128
#define Sn 4
#define BVn (Bn * Vn)          // 409600 rows, divisible by 16
#define EPSV 1e-5f

typedef __attribute__((ext_vector_type(2))) float v2f;
typedef __attribute__((ext_vector_type(8))) float v8f;
typedef __attribute__((ext_vector_type(4))) unsigned int u32x4;
typedef __attribute__((ext_vector_type(4))) int i32x4;
typedef __attribute__((ext_vector_type(8))) int i32x8;

#if defined(__has_builtin)
#if __has_builtin(__builtin_amdgcn_tensor_load_to_lds)
#define HAVE_TDM 1
#endif
#endif
#if defined(__has_include)
#if __has_include(<hip/amd_detail/amd_gfx1250_TDM.h>)
#define TDM_ARITY 6
#else
#define TDM_ARITY 5
#endif
#else
#define TDM_ARITY 5
#endif

#if defined(HAVE_TDM)
// One 2-D TDM tile load: global (td0 x td1 elements, row stride stride0, 4-byte
// elements) -> LDS at lds_addr. g1w0_extra carries pad_enable/interval/amount.
__device__ __forceinline__ void tdm_load_2d_f32(
    unsigned lds_addr, const void* gp, unsigned td0, unsigned td1,
    unsigned tile0, unsigned tile1, unsigned stride0, unsigned g1w0_extra)
{
    unsigned long long ga = (unsigned long long)(size_t)gp;
    u32x4 g0 = {0u, 0u, 0u, 0u};
    g0.x = 1u;                                   // count=1 (valid), is_restore=0
    g0.y = lds_addr;                             // lds_addr [63:32]
    g0.z = (unsigned)ga;                         // global_addr low
    g0.w = (unsigned)((ga >> 32) & 0x01ffffffu)  // global_addr [56:32]
           | (2u << 30);                         // type=2 ("image")
    i32x8 g1 = {0, 0, 0, 0, 0, 0, 0, 0};
    g1[0] = (int)((2u << 16) | g1w0_extra);      // data_size=4B (+ pad cfg)
    g1[1] = (int)((td0 & 0xffffu) << 16);                      // tensor_dim0 lo
    g1[2] = (int)(((td0 >> 16) & 0xffffu) | ((td1 & 0xffffu) << 16));
    g1[3] = (int)(((td1 >> 16) & 0xffffu) | ((tile0 & 0xffffu) << 16));
    g1[4] = (int)(tile1 & 0xffffu);              // tile_dim1 (tile_dim2=0)
    g1[5] = (int)stride0;                        // tensor_dim0_stride low 32
    i32x4 z4 = {0, 0, 0, 0};
#if TDM_ARITY == 6
    i32x8 z8 = {0, 0, 0, 0, 0, 0, 0, 0};
    __builtin_amdgcn_tensor_load_to_lds(g0, g1, z4, z4, z8, 0);
#else
    __builtin_amdgcn_tensor_load_to_lds(g0, g1, z4, z4, 0);
#endif
}
#endif

// ---------------- ws layout (floats) ----------------
// [0,512)      sum1[s*C+c]
// [512,1024)   sq1[s*C+c]
// [1024,1152)  sum2[c]
// [1152,1280)  sq2[c]
// [2048, +S*BV*C)   h buffer
// then BV*C          m_pre buffer

__global__ __launch_bounds__(256) void msg3d_zero_stats(float* ws) {
    int t = threadIdx.x;
    for (int i = t; i < 1280; i += 256) ws[i] = 0.0f;
}

// ---------------------------------------------------------------------------
// Kernel 1: h[s] = (A[s] @ x) @ W[s] + b[s]; accumulate per-(s,c) sum/sumsq
// block = 128 threads (4 waves), tile = 16 rows x 128 cols, grid = (BV/16, S)
// W[s] and the x rows are staged into LDS by the Tensor Data Mover; the
// 136-float W pitch is produced by TDM padding (8 DWORDs after each 128).
// ---------------------------------------------------------------------------
__global__ __launch_bounds__(128) void msg3d_gemm1(
    const float* __restrict__ x, const float* __restrict__ A,
    const float* __restrict__ W, const float* __restrict__ bias,
    float* __restrict__ hbuf, float* __restrict__ sum1, float* __restrict__ sq1)
{
    __shared__ float Abuf[Vn * Vn];        // 625
    __shared__ float Wl[Cn * 136];         // padded pitch 136 (conflict-free B reads)
    __shared__ float xl[2 * Vn * Cn];      // up to 2 samples of x
    __shared__ float aggl[16 * 132];       // padded pitch 132 (conflict-free A reads)
    __shared__ float ssum[Cn], ssq[Cn];

    const int s   = blockIdx.y;
    const int m0  = blockIdx.x * 16;
    const int tid = threadIdx.x;

    const int b0 = m0 / Vn;
    const int b1 = (m0 + 15) / Vn;
    const int nsamp = b1 - b0 + 1;

    for (int i = tid; i < Vn * Vn; i += 128) Abuf[i] = A[s * Vn * Vn + i];

#if defined(HAVE_TDM)
    if (tid == 0) {
        // W[s]: 128x128 fp32 tile, LDS pitch padded to 136 floats:
        // pad_enable(bit20), pad_interval=6 (128 DWORDs), pad_amount=7 (8 DWORDs)
        tdm_load_2d_f32((unsigned)(size_t)(void*)Wl,
                        W + (size_t)s * Cn * Cn,
                        Cn, Cn, Cn, Cn, Cn,
                        (1u << 20) | (6u << 22) | (7u << 25));
        // x rows for the touched samples: contiguous run, 1 row of nf elements
        const unsigned nf = (unsigned)(nsamp * Vn * Cn);
        tdm_load_2d_f32((unsigned)(size_t)(void*)xl,
                        x + (size_t)b0 * Vn * Cn,
                        nf, 1u, nf, 1u, nf, 0u);
    }
    __builtin_amdgcn_s_wait_tensorcnt((short)0);
#else
    { // W[s] -> LDS (padded pitch), float4 loads
        const float4* Wg = (const float4*)(W + (size_t)s * Cn * Cn);
        for (int i4 = tid; i4 < Cn * Cn / 4; i4 += 128) {
            int k = i4 >> 5, n4 = (i4 & 31) * 4;
            *(float4*)&Wl[k * 136 + n4] = Wg[i4];
        }
    }
    { // x rows for the touched samples -> LDS
        const float4* xg = (const float4*)(x + (size_t)b0 * Vn * Cn);
        float4* xl4 = (float4*)xl;
        const int nf4 = nsamp * Vn * Cn / 4;
        for (int i = tid; i < nf4; i += 128) xl4[i] = xg[i];
    }
#endif
    if (tid < Cn) { ssum[tid] = 0.0f; ssq[tid] = 0.0f; }
    __syncthreads();

    // Graph aggregation: agg[m][c] = sum_u A[s][v_m][u] * x[b_m][u][c]
    {
        const int m  = tid >> 3;
        const int c0 = (tid & 7) * 16;
        const int row = m0 + m;
        const int b = row / Vn, v = row % Vn, lb = b - b0;
        float accv[16];
#pragma unroll
        for (int e = 0; e < 16; ++e) accv[e] = 0.0f;
        const float* Arow = &Abuf[v * Vn];
        for (int u = 0; u < Vn; ++u) {
            const float a = Arow[u];
            const float* xr = &xl[(lb * Vn + u) * Cn + c0];
#pragma unroll
            for (int e = 0; e < 16; ++e) accv[e] = fmaf(a, xr[e], accv[e]);
        }
#pragma unroll
        for (int e = 0; e < 16; ++e) aggl[m * 132 + c0 + e] = accv[e];
    }
    __syncthreads();

    const int lane  = tid & 31;
    const int wave  = tid >> 5;
    const int lm    = lane & 15;
    const int khalf = lane >> 4;   // WMMA f32: VGPR j holds K = j + 2*khalf

#pragma unroll
    for (int t = 0; t < 2; ++t) {
        const int nt = wave * 2 + t;
        const int n  = nt * 16 + lm;
        v8f acc = {};
        for (int kt = 0; kt < Cn / 4; ++kt) {
            const int k0 = kt * 4 + 2 * khalf;
            v2f av, bv;
            av.x = aggl[lm * 132 + k0];
            av.y = aggl[lm * 132 + k0 + 1];
            bv.x = Wl[k0 * 136 + n];
            bv.y = Wl[(k0 + 1) * 136 + n];
            acc = __builtin_amdgcn_wmma_f32_16x16x4_f32(
                false, av, false, bv, (short)0, acc, false, false);
        }
        const float bb = bias[s * Cn + n];
        float ls = 0.0f, lq = 0.0f;
        const size_t base = (size_t)s * BVn * Cn;
#pragma unroll
        for (int i = 0; i < 8; ++i) {
            const int m = i + 8 * khalf;
            const float hv = acc[i] + bb;
            ls += hv; lq += hv * hv;
            hbuf[base + (size_t)(m0 + m) * Cn + n] = hv;
        }
        atomicAdd(&ssum[n], ls);
        atomicAdd(&ssq[n], lq);
    }
    __syncthreads();
    if (tid < Cn) {
        atomicAdd(&sum1[s * Cn + tid], ssum[tid]);
        atomicAdd(&sq1[s * Cn + tid], ssq[tid]);
    }
}

// ---------------------------------------------------------------------------
// Kernel 2: m_pre = relu(BN1(h)) concat @ Wm + bm; accumulate BN2 stats
// block = 128 threads, tile = 16 rows x 128 cols, K = 512
// ---------------------------------------------------------------------------
__global__ __launch_bounds__(128) void msg3d_gemm2(
    const float* __restrict__ hbuf, const float* __restrict__ Wm,
    const float* __restrict__ bm, const float* __restrict__ gamma,
    const float* __restrict__ beta, const float* __restrict__ sum1,
    const float* __restrict__ sq1, float* __restrict__ mbuf,
    float* __restrict__ sum2, float* __restrict__ sq2)
{
    __shared__ float scl[Sn * Cn], shf[Sn * Cn];
    __shared__ float al[16 * 516];         // padded pitch 516
    __shared__ float ssum[Cn], ssq[Cn];

    const int m0  = blockIdx.x * 16;
    const int tid = threadIdx.x;
    const float invN = 1.0f / (float)BVn;

    for (int i = tid; i < Sn * Cn; i += 128) {
        const float mu  = sum1[i] * invN;
        const float var = sq1[i] * invN - mu * mu;
        const float g   = gamma[i] * rsqrtf(var + EPSV);
        scl[i] = g;
        shf[i] = beta[i] - mu * g;
    }
    if (tid < Cn) { ssum[tid] = 0.0f; ssq[tid] = 0.0f; }
    __syncthreads();

    // Stage relu(BN1(h)) rows: kk = s*128+c
    for (int i = tid; i < 16 * 512; i += 128) {
        const int m  = i >> 9;
        const int kk = i & 511;
        const int s = kk >> 7, c = kk & 127;
        float v = hbuf[(size_t)s * BVn * Cn + (size_t)(m0 + m) * Cn + c];
        v = fmaf(v, scl[kk], shf[kk]);
        al[m * 516 + kk] = fmaxf(v, 0.0f);
    }
    __syncthreads();

    const int lane  = tid & 31;
    const int wave  = tid >> 5;
    const int lm    = lane & 15;
    const int khalf = lane >> 4;

#pragma unroll
    for (int t = 0; t < 2; ++t) {
        const int nt = wave * 2 + t;
        const int n  = nt * 16 + lm;
        v8f acc = {};
        for (int kt = 0; kt < 512 / 4; ++kt) {
            const int k0 = kt * 4 + 2 * khalf;
            v2f av, bv;
            av.x = al[lm * 516 + k0];
            av.y = al[lm * 516 + k0 + 1];
            bv.x = Wm[k0 * Cn + n];          // Wm is L2-resident (256 KB)
            bv.y = Wm[(k0 + 1) * Cn + n];
            acc = __builtin_amdgcn_wmma_f32_16x16x4_f32(
                false, av, false, bv, (short)0, acc, false, false);
        }
        const float bb = bm[n];
        float ls = 0.0f, lq = 0.0f;
#pragma unroll
        for (int i = 0; i < 8; ++i) {
            const int m = i + 8 * khalf;
            const float mv = acc[i] + bb;
            ls += mv; lq += mv * mv;
            mbuf[(size_t)(m0 + m) * Cn + n] = mv;
        }
        atomicAdd(&ssum[n], ls);
        atomicAdd(&ssq[n], lq);
    }
    __syncthreads();
    if (tid < Cn) {
        atomicAdd(&sum2[tid], ssum[tid]);
        atomicAdd(&sq2[tid], ssq[tid]);
    }
}

// ---------------------------------------------------------------------------
// Kernel 3: out = x + relu(BN2(m_pre))
// ---------------------------------------------------------------------------
__global__ __launch_bounds__(256) void msg3d_final(
    const float* __restrict__ x, const float* __restrict__ mbuf,
    const float* __restrict__ sum2, const float* __restrict__ sq2,
    const float* __restrict__ mg, const float* __restrict__ mbta,
    float* __restrict__ out)
{
    __shared__ float scl[Cn], shf[Cn];
    const int tid = threadIdx.x;
    if (tid < Cn) {
        const float invN = 1.0f / (float)BVn;
        const float mu  = sum2[tid] * invN;
        const float var = sq2[tid] * invN - mu * mu;
        const float g   = mg[tid] * rsqrtf(var + EPSV);
        scl[tid] = g;
        shf[tid] = mbta[tid] - mu * g;
    }
    __syncthreads();
    const float4* m4 = (const float4*)mbuf;
    const float4* x4 = (const float4*)x;
    float4* o4 = (float4*)out;
    const size_t total = (size_t)BVn * Cn / 4;
    for (size_t i = (size_t)blockIdx.x * blockDim.x + tid; i < total;
         i += (size_t)gridDim.x * blockDim.x) {
        const int c0 = (int)((i * 4) & (Cn - 1));
        const float4 mv = m4[i];
        const float4 xv = x4[i];
        float4 ov;
        ov.x = xv.x + fmaxf(fmaf(mv.x, scl[c0 + 0], shf[c0 + 0]), 0.0f);
        ov.y = xv.y + fmaxf(fmaf(mv.y, scl[c0 + 1], shf[c0 + 1]), 0.0f);
        ov.z = xv.z + fmaxf(fmaf(mv.z, scl[c0 + 2], shf[c0 + 2]), 0.0f);
        ov.w = xv.w + fmaxf(fmaf(mv.w, scl[c0 + 3], shf[c0 + 3]), 0.0f);
        o4[i] = ov;
    }
}

extern "C" void kernel_launch(void* const* d_in, const int* in_sizes, int n_in,
                              void* d_out, int out_size, void* d_ws, size_t ws_size,
                              hipStream_t stream) {
    (void)in_sizes; (void)n_in; (void)out_size; (void)ws_size;
    const float* x   = (const float*)d_in[0];
    const float* A   = (const float*)d_in[1];
    const float* W   = (const float*)d_in[2];
    const float* b   = (const float*)d_in[3];
    const float* g1  = (const float*)d_in[4];
    const float* be1 = (const float*)d_in[5];
    const float* Wm  = (const float*)d_in[6];
    const float* bm  = (const float*)d_in[7];
    const float* g2  = (const float*)d_in[8];
    const float* be2 = (const float*)d_in[9];
    float* ws   = (float*)d_ws;
    float* sum1 = ws;
    float* sq1  = ws + 512;
    float* sum2 = ws + 1024;
    float* sq2  = ws + 1152;
    float* hbuf = ws + 2048;
    float* mbuf = hbuf + (size_t)Sn * BVn * Cn;
    float* out  = (float*)d_out;

    msg3d_zero_stats<<<dim3(1), dim3(256), 0, stream>>>(ws);
    msg3d_gemm1<<<dim3(BVn / 16, Sn), dim3(128), 0, stream>>>(
        x, A, W, b, hbuf, sum1, sq1);
    msg3d_gemm2<<<dim3(BVn / 16), dim3(128), 0, stream>>>(
        hbuf, Wm, bm, g1, be1, sum1, sq1, mbuf, sum2, sq2);
    msg3d_final<<<dim3(4096), dim3(256), 0, stream>>>(
        x, mbuf, sum2, sq2, g2, be2, out);
}